// EpisodeReturns_5583457484848
// MI455X (gfx1250) — compile-verified
//
#include <hip/hip_runtime.h>

#define B_ROWS   4096
#define T_LEN    8192
#define GAMMA    0.99f
#define EPS_NORM 1e-9f

// --- CDNA5 async global->LDS path (gfx1250), with compile-safe fallback -----
typedef int v4i __attribute__((vector_size(16)));
typedef __attribute__((address_space(1))) v4i GV4i;   // global (AS1 == __device__)
typedef __attribute__((address_space(3))) v4i LV4i;   // LDS    (AS3)

#if __has_builtin(__builtin_amdgcn_global_load_async_to_lds_b128) && \
    __has_builtin(__builtin_amdgcn_s_wait_asynccnt)
#define HAVE_ASYNC_LDS 1
// lowers to: global_load_async_to_lds_b128 (tracked by ASYNCcnt)
#define ASYNC_COPY_B128(gp, lp) \
    __builtin_amdgcn_global_load_async_to_lds_b128((GV4i*)(gp), (LV4i*)(lp), 0, 0)
#define WAIT_ASYNC(n) __builtin_amdgcn_s_wait_asynccnt(n)
#else
#define HAVE_ASYNC_LDS 0
#define ASYNC_COPY_B128(gp, lp) (*(lp) = *(gp))   // sync global->VGPR->ds_store
#define WAIT_ASYNC(n) ((void)0)
#endif

// ---------------------------------------------------------------------------
// Pass 1: reverse-time discounted-return scan, one wave32 per row.
// Chunk = 128 elements (lane -> one float4). 3-deep async pipeline: chunks
// c-1 and c-2 staged into LDS by the TDM-side async path while chunk c is
// scanned; consumption gated by s_wait_asynccnt. Per-lane serial compose into
// (A,B) with r_first = A*carry + B, 5-step wave suffix scan via shfl_down,
// per-element FMA reconstruction. Row sum/sumsq fused.
// ---------------------------------------------------------------------------
__global__ __launch_bounds__(256) void er_scan_kernel(
    const float* __restrict__ rew, const float* __restrict__ don,
    float* __restrict__ out,
    float* __restrict__ rowSum, float* __restrict__ rowSumSq)
{
    // per-wave private triple buffers: 8 waves x 3 x 32 float4 x 2 arrays = 24 KB
    __shared__ float4 sR[8][3][32];
    __shared__ float4 sD[8][3][32];

    const int lane = threadIdx.x & 31;
    const int wave = threadIdx.x >> 5;
    const int row  = blockIdx.x * 8 + wave;          // grid = B_ROWS/8 blocks

    const float4* __restrict__ r4 = (const float4*)rew + (size_t)row * (T_LEN / 4);
    const float4* __restrict__ d4 = (const float4*)don + (size_t)row * (T_LEN / 4);
    float4*       __restrict__ o4 = (float4*)out       + (size_t)row * (T_LEN / 4);

    const int NCHUNK = T_LEN / 128;                  // 64 chunks per row

    // prime: issue async stage-in for the two last-in-time chunks
    {
        int c = NCHUNK - 1, j = c * 32 + lane;
        ASYNC_COPY_B128(r4 + j, &sR[wave][c % 3][lane]);
        ASYNC_COPY_B128(d4 + j, &sD[wave][c % 3][lane]);
        c = NCHUNK - 2; j = c * 32 + lane;
        ASYNC_COPY_B128(r4 + j, &sR[wave][c % 3][lane]);
        ASYNC_COPY_B128(d4 + j, &sD[wave][c % 3][lane]);
    }

    float carry = 0.0f;                              // return at chunk-end time
    float sum = 0.0f, sumsq = 0.0f;

    for (int c = NCHUNK - 1; c >= 0; --c) {
        // keep two chunks in flight; L2 warm-up two more ahead
        if (c >= 2) {
            int j = (c - 2) * 32 + lane;
            ASYNC_COPY_B128(r4 + j, &sR[wave][(c - 2) % 3][lane]);
            ASYNC_COPY_B128(d4 + j, &sD[wave][(c - 2) % 3][lane]);
            if (c >= 4) {
                int p = (c - 4) * 32 + lane;
                __builtin_prefetch(&r4[p], 0, 3);    // -> global_prefetch_b8
                __builtin_prefetch(&d4[p], 0, 3);
            }
            WAIT_ASYNC(4);                           // retire chunk c's 2 loads
        } else if (c == 1) {
            WAIT_ASYNC(2);
        } else {
            WAIT_ASYNC(0);
        }

        float4 rc = sR[wave][c % 3][lane];           // ds_load_b128
        float4 dc = sD[wave][c % 3][lane];

        // per-element decay a_k = gamma*(1-done_k)
        float a0 = GAMMA * (1.0f - dc.x);
        float a1 = GAMMA * (1.0f - dc.y);
        float a2 = GAMMA * (1.0f - dc.z);
        float a3 = GAMMA * (1.0f - dc.w);

        // serial reverse compose of this lane's 4 elements: r_first = A*carry + B
        float A  = a3;
        float Bv = rc.w;                             // k=3
        Bv = fmaf(a2, Bv, rc.z); A *= a2;            // k=2
        Bv = fmaf(a1, Bv, rc.y); A *= a1;            // k=1
        Bv = fmaf(a0, Bv, rc.x); A *= a0;            // k=0

        // inclusive suffix scan across lanes (lane i aggregates segments i..31)
        float Ai = A, Bi = Bv;
        #pragma unroll
        for (int d = 1; d < 32; d <<= 1) {
            float A2 = __shfl_down(Ai, d, 32);
            float B2 = __shfl_down(Bi, d, 32);
            if (lane + d < 32) {
                Bi = fmaf(Ai, B2, Bi);               // earlier ∘ later
                Ai = Ai * A2;
            }
        }

        // exclusive suffix (from lane+1) -> this lane's incoming carry
        float Ae = __shfl_down(Ai, 1, 32);
        float Be = __shfl_down(Bi, 1, 32);
        if (lane == 31) { Ae = 1.0f; Be = 0.0f; }
        float tc = fmaf(Ae, carry, Be);

        // carry for the next (earlier) chunk = return at lane0/elem0
        float newCarry = __shfl(fmaf(Ai, carry, Bi), 0, 32);

        // reconstruct per-element returns (exact recurrence, fp32 FMA)
        float r3 = fmaf(a3, tc, rc.w);
        float r2 = fmaf(a2, r3, rc.z);
        float r1 = fmaf(a1, r2, rc.y);
        float r0 = fmaf(a0, r1, rc.x);

        float4 o; o.x = r0; o.y = r1; o.z = r2; o.w = r3;
        o4[c * 32 + lane] = o;                       // global_store_b128

        sum   += r0 + r1 + r2 + r3;
        sumsq += fmaf(r0, r0, fmaf(r1, r1, fmaf(r2, r2, r3 * r3)));
        carry  = newCarry;
    }

    // wave reduction of row statistics
    #pragma unroll
    for (int d = 16; d >= 1; d >>= 1) {
        sum   += __shfl_down(sum,   d, 32);
        sumsq += __shfl_down(sumsq, d, 32);
    }
    if (lane == 0) {
        rowSum[row]   = sum;
        rowSumSq[row] = sumsq;
    }
}

// ---------------------------------------------------------------------------
// Pass 2: single block reduces 4096 row sums -> global mean scalar.
// ---------------------------------------------------------------------------
__global__ __launch_bounds__(256) void er_mean_kernel(
    const float* __restrict__ rowSum, float* __restrict__ meanOut)
{
    __shared__ float sdata[256];
    float s = 0.0f;
    for (int i = threadIdx.x; i < B_ROWS; i += 256) s += rowSum[i];
    sdata[threadIdx.x] = s;
    __syncthreads();
    #pragma unroll
    for (int stride = 128; stride > 0; stride >>= 1) {
        if (threadIdx.x < stride) sdata[threadIdx.x] += sdata[threadIdx.x + stride];
        __syncthreads();
    }
    if (threadIdx.x == 0)
        meanOut[0] = sdata[0] / (float)((size_t)B_ROWS * (size_t)T_LEN);
}

// ---------------------------------------------------------------------------
// Pass 3: in-place normalize: (ret - global_mean) / (row_std_ddof1 + eps).
// ---------------------------------------------------------------------------
__global__ __launch_bounds__(256) void er_norm_kernel(
    float* __restrict__ out,
    const float* __restrict__ rowSum, const float* __restrict__ rowSumSq,
    const float* __restrict__ meanPtr)
{
    const int Q = T_LEN / 4;                         // float4s per row
    int idx4 = blockIdx.x * 256 + threadIdx.x;       // grid covers B*T/4
    int row  = idx4 / Q;

    float mean_g = meanPtr[0];
    float s  = rowSum[row];
    float ss = rowSumSq[row];
    float var = (ss - s * s * (1.0f / (float)T_LEN)) * (1.0f / (float)(T_LEN - 1));
    var = var > 0.0f ? var : 0.0f;
    float inv = 1.0f / (__builtin_sqrtf(var) + EPS_NORM);

    float4* o4 = (float4*)out;
    float4 v = o4[idx4];
    v.x = (v.x - mean_g) * inv;
    v.y = (v.y - mean_g) * inv;
    v.z = (v.z - mean_g) * inv;
    v.w = (v.w - mean_g) * inv;
    o4[idx4] = v;
}

// ---------------------------------------------------------------------------
extern "C" void kernel_launch(void* const* d_in, const int* in_sizes, int n_in,
                              void* d_out, int out_size, void* d_ws, size_t ws_size,
                              hipStream_t stream) {
    (void)in_sizes; (void)n_in; (void)out_size; (void)ws_size;
    const float* rewards = (const float*)d_in[0];
    const float* dones   = (const float*)d_in[1];
    float* out = (float*)d_out;

    float* ws       = (float*)d_ws;
    float* rowSum   = ws;                 // [B_ROWS]
    float* rowSumSq = ws + B_ROWS;        // [B_ROWS]
    float* meanPtr  = ws + 2 * B_ROWS;    // [1]

    er_scan_kernel<<<B_ROWS / 8, 256, 0, stream>>>(rewards, dones, out,
                                                   rowSum, rowSumSq);
    er_mean_kernel<<<1, 256, 0, stream>>>(rowSum, meanPtr);
    const int total4 = B_ROWS * (T_LEN / 4);
    er_norm_kernel<<<total4 / 256, 256, 0, stream>>>(out, rowSum, rowSumSq,
                                                     meanPtr);
}